// General_BEV_Query_Initialization_9543417332324
// MI455X (gfx1250) — compile-verified
//
#include <hip/hip_runtime.h>
#include <hip/hip_bf16.h>
#include <math.h>

// ---------------------------------------------------------------------------
// Problem constants
// ---------------------------------------------------------------------------
#define BATCH 4
#define CH 256
#define HH 180
#define WW 180
#define HWP (HH * WW)          // 32400
#define NC 10
#define NPROP 200
#define BN_EPS 1e-5f

// Use the gfx1250 async global->LDS engine for the (always in-bounds) weight
// staging in conv1. Flip to 0 if the assembler rejects the mnemonic.
#define ASYNC_WEIGHTS 1

// ---------------------------------------------------------------------------
// Output layout (floats, concatenated in reference return order)
// ---------------------------------------------------------------------------
#define OUT_QF 0                                   // [4,256,200]
#define OUT_QP (OUT_QF + BATCH * CH * NPROP)       // [4,200,2]
#define OUT_QL (OUT_QP + BATCH * NPROP * 2)        // [4,200]
#define OUT_QS (OUT_QL + BATCH * NPROP)            // [4,10,200]
#define OUT_DH (OUT_QS + BATCH * NC * NPROP)       // [4,10,180,180]

// ---------------------------------------------------------------------------
// Workspace layout (byte offsets)
// ---------------------------------------------------------------------------
#define WS_X1    0ull                                          // [4,256,180,180] f32
#define WS_X1_SZ (4ull * CH * HH * WW * 4ull)
#define WS_HEAT  (WS_X1 + WS_X1_SZ)                            // sigmoid heatmap
#define WS_HM_SZ (4ull * NC * HWP * 4ull)
#define WS_NMS   (WS_HEAT + WS_HM_SZ)                          // masked heatmap
#define WS_THR   (WS_NMS + WS_HM_SZ)                           // 4 x u32 thresholds
#define WS_CNTA  (WS_THR + 16ull)                              // 4 x u32
#define WS_CNTB  (WS_CNTA + 16ull)                             // 4 x u32
#define WS_GV    (WS_CNTB + 16ull)                             // 4*256 u32 (values > T)
#define WS_GI    (WS_GV + 4ull * 256 * 4)                      // 4*256 u32 (their idx)
#define WS_EQ    (WS_GI + 4ull * 256 * 4)                      // 4*1024 u32 (idx == T)
#define WS_TOP   (WS_EQ + 4ull * 1024 * 4)                     // 4*200 u32 sorted

typedef float v2f __attribute__((ext_vector_type(2)));
typedef float v8f __attribute__((ext_vector_type(8)));

__device__ __forceinline__ v8f wmma_f32_k4(v2f a, v2f b, v8f c) {
  // D = A(16x4 f32) * B(4x16 f32) + C(16x16 f32)
  return __builtin_amdgcn_wmma_f32_16x16x4_f32(false, a, false, b, (short)0, c,
                                               false, false);
}

#if ASYNC_WEIGHTS
__device__ __forceinline__ void async_g2l_b32(const float* gsrc, float* lds_dst) {
  // GLOBAL_LOAD_ASYNC_TO_LDS_B32: VDST = LDS byte address (low 32 bits of the
  // generic shared pointer == wave-relative LDS address per flat-aperture
  // truncation), VADDR = 64-bit global address, SADDR = off. Tracked by
  // ASYNCcnt.
  unsigned lds_addr = (unsigned)(unsigned long long)lds_dst;
  asm volatile("global_load_async_to_lds_b32 %0, %1, off"
               :: "v"(lds_addr), "v"(gsrc) : "memory");
}
__device__ __forceinline__ void wait_asynccnt0() {
  asm volatile("s_wait_asynccnt 0x0" ::: "memory");
}
#endif

// ===========================================================================
// Kernel 1: conv1 3x3 (256->256, no bias) + BN(running stats) + ReLU -> x1
// Implicit GEMM: M = 64 pixels along x, N = 256 out channels, K = 9*256.
// grid (3, 180, 4), block 256 (8 waves); wave w owns channels [w*32, w*32+32)
// LDS tiles are channel-minor (pad 16->18) so every A/B fragment is one
// aligned ds_load_b64 and 16-lane accesses are bank-conflict-free.
// ===========================================================================
__global__ __launch_bounds__(256)
void conv1_bn_relu_wmma(const float* __restrict__ in,     // [4,256,180,180]
                        const float* __restrict__ w,      // [256,256,3,3]
                        const float* __restrict__ gamma,
                        const float* __restrict__ beta,
                        const float* __restrict__ mean,
                        const float* __restrict__ var,
                        float* __restrict__ x1) {
  const int b  = blockIdx.z;
  const int y  = blockIdx.y;
  const int x0 = blockIdx.x * 64;
  const int tid  = threadIdx.x;
  const int wave = tid >> 5;
  const int lane = tid & 31;
  const int row   = lane & 15;   // M (A) / N (B,C,D) coordinate
  const int khalf = lane >> 4;   // K-half selector for A/B fragments
  const int n_base = wave * 32;

  __shared__ float lin[68][18];        // [col j][in-ch]; col j -> x = x0-1+j
  __shared__ float lw[3][256][18];     // [kx][out-ch][in-ch]

  v8f acc[4][2];
#pragma unroll
  for (int mt = 0; mt < 4; ++mt)
#pragma unroll
    for (int nt = 0; nt < 2; ++nt) acc[mt][nt] = v8f{0.f,0.f,0.f,0.f,0.f,0.f,0.f,0.f};

  for (int cb = 0; cb < CH; cb += 16) {
    for (int ky = 0; ky < 3; ++ky) {
      __syncthreads();
      const int yy = y + ky - 1;
      // stage input strip: 66 cols x 16 channels (channel-minor)
      for (int i = tid; i < 66 * 16; i += 256) {
        const int c = i & 15;
        const int j = i >> 4;
        const int xx = x0 - 1 + j;
        float v = 0.f;
        if (yy >= 0 && yy < HH && xx >= 0 && xx < WW)
          v = in[(((size_t)b * CH + cb + c) * HH + yy) * WW + xx];
        lin[j][c] = v;
      }
      // prefetch next channel block's row into cache (global_prefetch_b8)
      if (cb + 16 < CH && yy >= 0 && yy < HH) {
        const int c = tid & 15;
        const int j = (tid >> 4) * 4;
        const int xx = x0 - 1 + j;
        if (j < 66 && xx >= 0 && xx < WW)
          __builtin_prefetch(&in[(((size_t)b * CH + cb + 16 + c) * HH + yy) * WW + xx], 0, 0);
      }
      // stage weights w[o, cb+c, ky, kx] -> lw[kx][o][c]
#if ASYNC_WEIGHTS
      for (int i = tid; i < 3 * 16 * 256; i += 256) {
        const int o  = i & 255;
        const int c  = (i >> 8) & 15;
        const int kx = i >> 12;
        async_g2l_b32(&w[(((size_t)o * CH + cb + c) * 3 + ky) * 3 + kx],
                      &lw[kx][o][c]);
      }
      wait_asynccnt0();
#else
      for (int i = tid; i < 3 * 16 * 256; i += 256) {
        const int o  = i & 255;
        const int c  = (i >> 8) & 15;
        const int kx = i >> 12;
        lw[kx][o][c] = w[(((size_t)o * CH + cb + c) * 3 + ky) * 3 + kx];
      }
#endif
      __syncthreads();

#pragma unroll
      for (int kx = 0; kx < 3; ++kx) {
#pragma unroll
        for (int k4 = 0; k4 < 4; ++k4) {
          const int kc = k4 * 4 + khalf * 2;   // channel pair within 16-block
          v2f bfr[2];
#pragma unroll
          for (int nt = 0; nt < 2; ++nt) {
            const int n = n_base + nt * 16 + row;
            bfr[nt] = *(const v2f*)&lw[kx][n][kc];        // one ds_load_b64
          }
#pragma unroll
          for (int mt = 0; mt < 4; ++mt) {
            const int j = mt * 16 + row + kx;
            v2f a = *(const v2f*)&lin[j][kc];             // one ds_load_b64
#pragma unroll
            for (int nt = 0; nt < 2; ++nt)
              acc[mt][nt] = wmma_f32_k4(a, bfr[nt], acc[mt][nt]);
          }
        }
      }
    }
  }

  // Epilogue: BN + ReLU, store x1. D element r: pixel m = mt*16+r+8*khalf,
  // channel = n_base + nt*16 + row.
#pragma unroll
  for (int nt = 0; nt < 2; ++nt) {
    const int ch = n_base + nt * 16 + row;
    const float sc = gamma[ch] * rsqrtf(var[ch] + BN_EPS);
    const float sh = beta[ch] - mean[ch] * sc;
#pragma unroll
    for (int mt = 0; mt < 4; ++mt) {
#pragma unroll
      for (int r = 0; r < 8; ++r) {
        const int x = x0 + mt * 16 + r + 8 * khalf;
        if (x < WW) {
          const float v = acc[mt][nt][r] * sc + sh;
          x1[(((size_t)b * CH + ch) * HH + y) * WW + x] = fmaxf(v, 0.f);
        }
      }
    }
  }
}

// ===========================================================================
// Kernel 2: conv2 3x3 (256->10) + bias; writes dense_heatmap (output) and
// sigmoid heatmap (workspace). grid (1, 180, 4), block 128 (4 waves);
// wave w covers pixels [w*48, w*48+48), N padded 10->16.
// ===========================================================================
__global__ __launch_bounds__(128)
void conv2_sigmoid_wmma(const float* __restrict__ x1,
                        const float* __restrict__ w2,   // [10,256,3,3]
                        const float* __restrict__ b2,   // [10]
                        float* __restrict__ dh,         // d_out + OUT_DH
                        float* __restrict__ heat) {
  const int b = blockIdx.z;
  const int y = blockIdx.y;
  const int tid  = threadIdx.x;
  const int wave = tid >> 5;
  const int lane = tid & 31;
  const int row   = lane & 15;
  const int khalf = lane >> 4;
  const int wbase = wave * 48;

  __shared__ float lin2[196][18];      // [col j][in-ch]; col j -> x = j-1
  __shared__ float lw2[3][16][18];     // [kx][out n (10 used)][in-ch]

  v8f acc[3];
#pragma unroll
  for (int mt = 0; mt < 3; ++mt) acc[mt] = v8f{0.f,0.f,0.f,0.f,0.f,0.f,0.f,0.f};

  for (int cb = 0; cb < CH; cb += 16) {
    for (int ky = 0; ky < 3; ++ky) {
      __syncthreads();
      const int yy = y + ky - 1;
      for (int i = tid; i < 196 * 16; i += 128) {
        const int c = i & 15;
        const int j = i >> 4;
        const int xx = j - 1;
        float v = 0.f;
        if (j >= 1 && j <= 180 && yy >= 0 && yy < HH && xx < WW)
          v = x1[(((size_t)b * CH + cb + c) * HH + yy) * WW + xx];
        lin2[j][c] = v;
      }
      for (int i = tid; i < 3 * 16 * 16; i += 128) {
        const int kx = i >> 8;
        const int c  = (i >> 4) & 15;
        const int n  = i & 15;
        lw2[kx][n][c] = (n < NC)
            ? w2[(((size_t)n * CH + cb + c) * 3 + ky) * 3 + kx] : 0.f;
      }
      __syncthreads();

#pragma unroll
      for (int kx = 0; kx < 3; ++kx) {
#pragma unroll
        for (int k4 = 0; k4 < 4; ++k4) {
          const int kc = k4 * 4 + khalf * 2;
          v2f bfr = *(const v2f*)&lw2[kx][row][kc];
#pragma unroll
          for (int mt = 0; mt < 3; ++mt) {
            const int j = wbase + mt * 16 + row + kx;
            v2f a = *(const v2f*)&lin2[j][kc];
            acc[mt] = wmma_f32_k4(a, bfr, acc[mt]);
          }
        }
      }
    }
  }

  const int ch = row;             // N coordinate == out class
  if (ch < NC) {
    const float bias = b2[ch];
#pragma unroll
    for (int mt = 0; mt < 3; ++mt) {
#pragma unroll
      for (int r = 0; r < 8; ++r) {
        const int x = wbase + mt * 16 + r + 8 * khalf;
        if (x < WW) {
          const float v = acc[mt][r] + bias;
          const size_t o = (((size_t)b * NC + ch) * HH + y) * WW + x;
          dh[o]   = v;
          heat[o] = 1.0f / (1.0f + expf(-v));
        }
      }
    }
  }
}

// ===========================================================================
// Kernel 3: NMS masking. classes 0-7: keep iff equal to 3x3 max (zero border),
// classes 8,9: keep always.
// ===========================================================================
__global__ __launch_bounds__(256)
void nms_kernel(const float* __restrict__ heat, float* __restrict__ nms) {
  const int idx = blockIdx.x * 256 + threadIdx.x;
  const int total = BATCH * NC * HWP;
  if (idx >= total) return;
  const int x  = idx % WW;
  const int y  = (idx / WW) % HH;
  const int ch = (idx / HWP) % NC;
  const float h = heat[idx];
  float keep;
  if (ch >= 8) {
    keep = h;                         // identity local max
  } else if (x == 0 || x == WW - 1 || y == 0 || y == HH - 1) {
    keep = (h == 0.f) ? h : 0.f;      // border local_max = 0
  } else {
    float m = -INFINITY;
#pragma unroll
    for (int dy = -1; dy <= 1; ++dy)
#pragma unroll
      for (int dx = -1; dx <= 1; ++dx)
        m = fmaxf(m, heat[idx + dy * WW + dx]);
    keep = (h == m) ? h : 0.f;
  }
  nms[idx] = keep;
}

// ===========================================================================
// Kernel 4: zero the atomic counters (fresh every launch; graph-replay safe)
// ===========================================================================
__global__ void zero_counters(unsigned* cntA, unsigned* cntB) {
  if (threadIdx.x < BATCH) { cntA[threadIdx.x] = 0u; cntB[threadIdx.x] = 0u; }
}

// ===========================================================================
// Kernel 5: per-batch exact 200th-largest via 31-step binary search over
// float bit patterns (all values >= 0 so uint order == float order).
// grid 4 x 1024 threads.
// ===========================================================================
__global__ __launch_bounds__(1024)
void select_threshold(const float* __restrict__ nms, unsigned* __restrict__ thr) {
  const int b   = blockIdx.x;
  const int tid = threadIdx.x;
  const unsigned* v = (const unsigned*)(nms + (size_t)b * NC * HWP);
  const int N = NC * HWP;
  __shared__ unsigned red;
  unsigned T = 0u;
  for (int bit = 30; bit >= 0; --bit) {   // bit 31 = sign, always 0
    const unsigned cand = T | (1u << bit);
    if (tid == 0) red = 0u;
    __syncthreads();
    unsigned c = 0;
    for (int i = tid; i < N; i += 1024) c += (v[i] >= cand) ? 1u : 0u;
    atomicAdd(&red, c);
    __syncthreads();
    const unsigned total = red;
    __syncthreads();
    if (total >= NPROP) T = cand;
  }
  if (tid == 0) thr[b] = T;
}

// ===========================================================================
// Kernel 6: compact candidates. Strictly-greater go to (gv,gi) (count < 200
// guaranteed); equal-to-threshold indices go to eq (cap 1024).
// grid (ceil(N/256), 4).
// ===========================================================================
__global__ __launch_bounds__(256)
void compact_candidates(const float* __restrict__ nms,
                        const unsigned* __restrict__ thr,
                        unsigned* __restrict__ cntA, unsigned* __restrict__ cntB,
                        unsigned* __restrict__ gv, unsigned* __restrict__ gi,
                        unsigned* __restrict__ eq) {
  const int b = blockIdx.y;
  const int i = blockIdx.x * 256 + threadIdx.x;
  const int N = NC * HWP;
  if (i >= N) return;
  const unsigned vb = ((const unsigned*)(nms + (size_t)b * N))[i];
  const unsigned T  = thr[b];
  if (vb > T) {
    const unsigned s = atomicAdd(&cntA[b], 1u);
    if (s < 256u) { gv[b * 256 + s] = vb; gi[b * 256 + s] = (unsigned)i; }
  } else if (vb == T) {
    const unsigned s = atomicAdd(&cntB[b], 1u);
    if (s < 1024u) eq[b * 1024 + s] = (unsigned)i;
  }
}

// ===========================================================================
// Kernel 7: order into exact top-k semantics (descending value, ties by
// ascending flat index -- matches stable jax.lax.top_k). grid 4 x 256.
// ===========================================================================
__global__ __launch_bounds__(256)
void order_topk(const unsigned* __restrict__ cntA, const unsigned* __restrict__ cntB,
                const unsigned* __restrict__ gv, const unsigned* __restrict__ gi,
                const unsigned* __restrict__ eq, unsigned* __restrict__ top) {
  const int b   = blockIdx.x;
  const int tid = threadIdx.x;
  __shared__ unsigned sgv[256], sgi[256], seq[1024];
  const int nA = min((int)cntA[b], NPROP);
  const int nB = min((int)cntB[b], 1024);
  for (int i = tid; i < nA; i += 256) { sgv[i] = gv[b * 256 + i]; sgi[i] = gi[b * 256 + i]; }
  for (int i = tid; i < nB; i += 256) seq[i] = eq[b * 1024 + i];
  __syncthreads();
  // strictly-greater set: full rank sort (value desc, index asc)
  for (int t = tid; t < nA; t += 256) {
    const unsigned v = sgv[t], ix = sgi[t];
    int rank = 0;
    for (int j = 0; j < nA; ++j)
      rank += (sgv[j] > v || (sgv[j] == v && sgi[j] < ix)) ? 1 : 0;
    top[b * NPROP + rank] = ix;
  }
  // equals fill the remainder in ascending-index order
  const int need = NPROP - nA;
  for (int t = tid; t < nB; t += 256) {
    const unsigned ix = seq[t];
    int r = 0;
    for (int j = 0; j < nB; ++j) r += (seq[j] < ix) ? 1 : 0;
    if (r < need) top[b * NPROP + nA + r] = ix;
  }
}

// ===========================================================================
// Kernel 8: gathers. One block per (b,p), 256 threads (one per channel).
// ===========================================================================
__global__ __launch_bounds__(256)
void gather_outputs(const float* __restrict__ bev,      // [4,256,HW]
                    const float* __restrict__ bev_pos,  // [1,HW,2]
                    const float* __restrict__ ce_w,     // [256,10,1]
                    const float* __restrict__ ce_b,     // [256]
                    const float* __restrict__ nms,      // [4,10,HW]
                    const unsigned* __restrict__ top,   // [4,200]
                    float* __restrict__ out) {
  const int bp = blockIdx.x;
  const int b  = bp / NPROP;
  const int p  = bp - b * NPROP;
  const int c  = threadIdx.x;
  const unsigned t = top[b * NPROP + p];
  const int label = (int)(t / HWP);
  const int pos   = (int)(t - (unsigned)label * HWP);
  // query_feat[b, c, p]
  out[OUT_QF + ((size_t)b * CH + c) * NPROP + p] =
      bev[((size_t)b * CH + c) * HWP + pos] + ce_w[c * NC + label] + ce_b[c];
  if (c < 2)
    out[OUT_QP + ((size_t)b * NPROP + p) * 2 + c] = bev_pos[(size_t)pos * 2 + c];
  if (c == 0)
    out[OUT_QL + (size_t)b * NPROP + p] = (float)label;
  if (c < NC)
    out[OUT_QS + ((size_t)b * NC + c) * NPROP + p] =
        nms[((size_t)b * NC + c) * HWP + pos];
}

// ===========================================================================
// Host entry
// ===========================================================================
extern "C" void kernel_launch(void* const* d_in, const int* in_sizes, int n_in,
                              void* d_out, int out_size, void* d_ws, size_t ws_size,
                              hipStream_t stream) {
  (void)in_sizes; (void)n_in; (void)out_size; (void)ws_size;
  const float* bev_feats = (const float*)d_in[0];
  const float* bev_pos   = (const float*)d_in[1];
  const float* conv1_w   = (const float*)d_in[2];
  const float* bn_gamma  = (const float*)d_in[3];
  const float* bn_beta   = (const float*)d_in[4];
  const float* bn_mean   = (const float*)d_in[5];
  const float* bn_var    = (const float*)d_in[6];
  const float* conv2_w   = (const float*)d_in[7];
  const float* conv2_b   = (const float*)d_in[8];
  const float* ce_w      = (const float*)d_in[9];
  const float* ce_b      = (const float*)d_in[10];

  float* out = (float*)d_out;
  char*  ws  = (char*)d_ws;
  float*    x1   = (float*)(ws + WS_X1);
  float*    heat = (float*)(ws + WS_HEAT);
  float*    nms  = (float*)(ws + WS_NMS);
  unsigned* thr  = (unsigned*)(ws + WS_THR);
  unsigned* cntA = (unsigned*)(ws + WS_CNTA);
  unsigned* cntB = (unsigned*)(ws + WS_CNTB);
  unsigned* gv   = (unsigned*)(ws + WS_GV);
  unsigned* gi   = (unsigned*)(ws + WS_GI);
  unsigned* eq   = (unsigned*)(ws + WS_EQ);
  unsigned* top  = (unsigned*)(ws + WS_TOP);

  // 1) conv1 + BN + ReLU (WMMA implicit GEMM, async weight staging)
  conv1_bn_relu_wmma<<<dim3(3, HH, BATCH), 256, 0, stream>>>(
      bev_feats, conv1_w, bn_gamma, bn_beta, bn_mean, bn_var, x1);

  // 2) conv2 + bias + sigmoid (WMMA)
  conv2_sigmoid_wmma<<<dim3(1, HH, BATCH), 128, 0, stream>>>(
      x1, conv2_w, conv2_b, out + OUT_DH, heat);

  // 3) NMS masking
  {
    const int total = BATCH * NC * HWP;
    nms_kernel<<<(total + 255) / 256, 256, 0, stream>>>(heat, nms);
  }

  // 4) reset counters
  zero_counters<<<1, 32, 0, stream>>>(cntA, cntB);

  // 5) per-batch 200th-largest threshold
  select_threshold<<<BATCH, 1024, 0, stream>>>(nms, thr);

  // 6) compact candidates
  {
    const int N = NC * HWP;
    compact_candidates<<<dim3((N + 255) / 256, BATCH), 256, 0, stream>>>(
        nms, thr, cntA, cntB, gv, gi, eq);
  }

  // 7) ordered top-200
  order_topk<<<BATCH, 256, 0, stream>>>(cntA, cntB, gv, gi, eq, top);

  // 8) gathers
  gather_outputs<<<BATCH * NPROP, 256, 0, stream>>>(
      bev_feats, bev_pos, ce_w, ce_b, nms, top, out);
}